// HuasdorffDisstanceLoss_73237782331755
// MI455X (gfx1250) — compile-verified
//
#include <hip/hip_runtime.h>
#include <stdint.h>

// Problem constants (from reference setup_inputs)
#define BB    4
#define CC    19
#define HH    256
#define WW    256
#define HWN   (HH * WW)          // 65536
#define NPIX  (BB * HWN)         // 262144
#define BIGD  512                // H + W, the reference's "big"
#define IGNORE_IDX 255

typedef __attribute__((ext_vector_type(2))) float v2f;
typedef __attribute__((ext_vector_type(8))) float v8f;

// ---------------------------------------------------------------------------
// Async global->LDS staging (CDNA5 GLOBAL_LOAD_ASYNC_TO_LDS_B32, ASYNCcnt).
// VDST carries the per-lane LDS byte offset; flat-aperture rule makes that the
// low 32 bits of the generic pointer. Completion enforced with
// s_wait_asynccnt 0 before the workgroup barrier.
// ---------------------------------------------------------------------------
__device__ __forceinline__ void async_load_f32_to_lds(const float* gptr, float* lptr) {
    unsigned lds_off = (unsigned)(uintptr_t)lptr;   // low 32 bits = LDS offset
    asm volatile("global_load_async_to_lds_b32 %0, %1, off"
                 :: "v"(lds_off), "v"(gptr)
                 : "memory");
}
__device__ __forceinline__ void wait_asynccnt0() {
    asm volatile("s_wait_asynccnt 0x0" ::: "memory");
}

// ---------------------------------------------------------------------------
// XOR-butterfly add step via ds_swizzle (group-of-32 mode):
// pattern = xor_mask << 10 | and_mask(0x1f). One DS op, no lane-bound guards.
// ---------------------------------------------------------------------------
template <int MASK>
__device__ __forceinline__ float xor_add_step(float x) {
    int yi = __builtin_amdgcn_ds_swizzle(__float_as_int(x), (MASK << 10) | 0x1f);
    return x + __int_as_float(yi);
}

// ---------------------------------------------------------------------------
// Wave-level sum via V_WMMA_F32_16X16X4_F32.
// A = ones(16x4) -> D[m][n] = colsum_n(B). B carries <lane value> in VGPR0 and
// 0 in VGPR1 (64 slots, 32 live). c[0] then holds colsum[n] (n = lane%16,
// duplicated in lanes 16-31). Summing all 16 column sums with a 4-step xor
// butterfly yields the exact f32 wave total, independent of B's element
// placement. EXEC is all ones at call sites (divergence-free 256-thread
// blocks).
// ---------------------------------------------------------------------------
__device__ __forceinline__ float wave_sum_wmma(float val) {
    v2f a; a[0] = 1.0f; a[1] = 1.0f;     // all-ones A (all 64 slots = 1.0)
    v2f b; b[0] = val;  b[1] = 0.0f;     // 32 live values + 32 zeros
    v8f c = {};
    c = __builtin_amdgcn_wmma_f32_16x16x4_f32(
            /*neg_a=*/false, a, /*neg_b=*/false, b,
            /*c_mod=*/(short)0, c, /*reuse_a=*/false, /*reuse_b=*/false);
    float s = c[0];
    s = xor_add_step<1>(s);
    s = xor_add_step<2>(s);
    s = xor_add_step<4>(s);
    s = xor_add_step<8>(s);
    return s;                             // every 16-group holds the wave total
}

// ---------------------------------------------------------------------------
// K1: softmax over channels, per-pixel delta sum, foreground masks.
//  S[p]     = sum_{c=1..18} (softmax_c(p) - float(tgt))^2
//  masks[p] = bit0: (tgt != 0), bit1: (argmax != 0)
// ---------------------------------------------------------------------------
__global__ void k1_softmax_delta(const float* __restrict__ pred,
                                 const int*   __restrict__ target,
                                 float*       __restrict__ S,
                                 unsigned char* __restrict__ masks) {
    int p = blockIdx.x * blockDim.x + threadIdx.x;
    if (p >= NPIX) return;
    int b  = p >> 16;          // / HWN
    int hw = p & 0xFFFF;       // % HWN
    const float* base = pred + (size_t)b * CC * HWN + hw;

    float v[CC];
    float mx = -__builtin_inff();
    int   am = 0;
    #pragma unroll
    for (int c = 0; c < CC; ++c) {
        float x = base[(size_t)c * HWN];
        v[c] = x;
        if (x > mx) { mx = x; am = c; }   // first-max, matches jnp.argmax
    }
    float sum = 0.0f;
    #pragma unroll
    for (int c = 0; c < CC; ++c) { v[c] = __expf(v[c] - mx); sum += v[c]; }
    float inv = 1.0f / sum;

    int t   = target[p];
    int tgt = (t != IGNORE_IDX) ? t : 0;
    float tf = (float)tgt;

    float s = 0.0f;
    #pragma unroll
    for (int c = 1; c < CC; ++c) {        // classes 1..C-1 only
        float d = v[c] * inv - tf;
        s = fmaf(d, d, s);
    }
    S[p] = s;
    masks[p] = (unsigned char)(((tgt != 0) ? 1 : 0) | ((am != 0) ? 2 : 0));
}

// ---------------------------------------------------------------------------
// K2: vertical EDT pass (phase 1 of Meijster) for both masks.
// One thread per (b, w) column; coalesced row-wise accesses.
// Reproduces reference edge semantics:
//   up = i + 512 if no background above, dn = 512 - i if none below,
//   g  = min(up, dn, 512); store g^2.
// ---------------------------------------------------------------------------
__global__ void k2_edt_vertical(const unsigned char* __restrict__ masks,
                                float* __restrict__ g2gt,
                                float* __restrict__ g2seg) {
    int tid = blockIdx.x * blockDim.x + threadIdx.x;   // 0 .. BB*WW-1
    if (tid >= BB * WW) return;
    int b = tid >> 8;
    int w = tid & (WW - 1);
    size_t colbase = (size_t)b * HWN + w;

    // pass 1 (top -> bottom): store "up" distance
    int lastg = -BIGD, lasts = -BIGD;
    for (int i = 0; i < HH; ++i) {
        size_t idx = colbase + (size_t)i * WW;
        unsigned char m = masks[idx];
        if (!(m & 1)) lastg = i;
        if (!(m & 2)) lasts = i;
        g2gt[idx]  = (float)(i - lastg);
        g2seg[idx] = (float)(i - lasts);
    }
    // pass 2 (bottom -> top): combine with "down" distance, cap, square
    int nextg = BIGD, nexts = BIGD;       // reference: dn = big - i when none
    for (int i = HH - 1; i >= 0; --i) {
        size_t idx = colbase + (size_t)i * WW;
        unsigned char m = masks[idx];
        if (!(m & 1)) nextg = i;
        if (!(m & 2)) nexts = i;
        float gg = fminf(fminf(g2gt[idx],  (float)(nextg - i)), (float)BIGD);
        float gs = fminf(fminf(g2seg[idx], (float)(nexts - i)), (float)BIGD);
        g2gt[idx]  = gg * gg;
        g2seg[idx] = gs * gs;
    }
}

// ---------------------------------------------------------------------------
// K3: horizontal EDT pass (min-plus over the row) + weighted accumulation.
// One 256-thread block per (b,h) row. g^2 rows staged into LDS with
// GLOBAL_LOAD_ASYNC_TO_LDS (no VGPR round-trip); the j' scan reads a uniform
// LDS address per iteration (conflict-free broadcast). Row sum produced with
// the WMMA wave reducer (deterministic tree).
// ---------------------------------------------------------------------------
__global__ void k3_phase2_reduce(const float* __restrict__ g2gt,
                                 const float* __restrict__ g2seg,
                                 const float* __restrict__ S,
                                 float*       __restrict__ rowsums) {
    __shared__ float sg[WW];
    __shared__ float ss[WW];
    __shared__ float wred[8];

    int row = blockIdx.x;                 // 0 .. BB*HH-1
    int j   = threadIdx.x;                // 0 .. 255
    size_t base = (size_t)row * WW;

    __builtin_prefetch(S + base + j, 0, 3);   // global_prefetch_b8

    // CDNA5 async DMA into LDS, tracked by ASYNCcnt
    async_load_f32_to_lds(g2gt  + base + j, &sg[j]);
    async_load_f32_to_lds(g2seg + base + j, &ss[j]);
    wait_asynccnt0();
    __syncthreads();

    float m1 = 1e30f, m2 = 1e30f;
    #pragma unroll 4
    for (int k = 0; k < WW; ++k) {
        float dx = (float)(j - k);
        float o  = dx * dx;
        m1 = fminf(m1, sg[k] + o);
        m2 = fminf(m2, ss[k] + o);
    }
    // dtm2 = seg_dtm^2 + gt_dtm^2 (sqrt-then-square in the reference is identity)
    float val = S[base + j] * (m1 + m2);

    float wsum = wave_sum_wmma(val);
    int lane = threadIdx.x & 31, wave = threadIdx.x >> 5;
    if (lane == 0) wred[wave] = wsum;
    __syncthreads();
    if (threadIdx.x == 0) {
        float s = 0.0f;
        #pragma unroll
        for (int i = 0; i < 8; ++i) s += wred[i];
        rowsums[row] = s;
    }
}

// ---------------------------------------------------------------------------
// K4: final deterministic reduction of BB*HH = 1024 row sums + scaling.
// loss = total / (B*H*W) / num_class  (LOSS_WEIGHT = 1)
// ---------------------------------------------------------------------------
__global__ void k4_final(const float* __restrict__ rowsums,
                         float*       __restrict__ out) {
    __shared__ float wred[8];
    int t = threadIdx.x;                  // 0 .. 255
    float v = rowsums[t] + rowsums[t + 256] + rowsums[t + 512] + rowsums[t + 768];
    float wsum = wave_sum_wmma(v);
    int lane = t & 31, wave = t >> 5;
    if (lane == 0) wred[wave] = wsum;
    __syncthreads();
    if (t == 0) {
        float tot = 0.0f;
        #pragma unroll
        for (int i = 0; i < 8; ++i) tot += wred[i];
        out[0] = tot * (float)(1.0 / ((double)NPIX * (double)CC));
    }
}

// ---------------------------------------------------------------------------
extern "C" void kernel_launch(void* const* d_in, const int* in_sizes, int n_in,
                              void* d_out, int out_size, void* d_ws, size_t ws_size,
                              hipStream_t stream) {
    const float* pred   = (const float*)d_in[0];   // (4,19,256,256) f32
    const int*   target = (const int*)d_in[1];     // (4,256,256) i32
    float* out = (float*)d_out;

    // Workspace layout (~3.4 MB total)
    float*         Sbuf   = (float*)d_ws;                         // NPIX f32
    unsigned char* mbuf   = (unsigned char*)(Sbuf + NPIX);        // NPIX u8
    float*         g2gt   = (float*)(mbuf + NPIX);                // NPIX f32
    float*         g2seg  = g2gt + NPIX;                          // NPIX f32
    float*         rowsum = g2seg + NPIX;                         // BB*HH f32

    (void)in_sizes; (void)n_in; (void)out_size; (void)ws_size;

    k1_softmax_delta<<<NPIX / 256, 256, 0, stream>>>(pred, target, Sbuf, mbuf);
    k2_edt_vertical<<<(BB * WW) / 256, 256, 0, stream>>>(mbuf, g2gt, g2seg);
    k3_phase2_reduce<<<BB * HH, 256, 0, stream>>>(g2gt, g2seg, Sbuf, rowsum);
    k4_final<<<1, 256, 0, stream>>>(rowsum, out);
}